// NeighborNorm_28819230556490
// MI455X (gfx1250) — compile-verified
//
#include <hip/hip_runtime.h>
#include <hip/hip_bf16.h>
#include <math.h>

#define DF 32
#define EPSF 1e-12f

// ---------------- CDNA5 feature detection (compile-safe) ----------------
#if __has_builtin(__builtin_amdgcn_wmma_f32_16x16x4_f32)
  #define HAVE_WMMA_F32 1
typedef __attribute__((ext_vector_type(2))) float v2f;
typedef __attribute__((ext_vector_type(8))) float v8f;
#endif

#if __has_builtin(__builtin_amdgcn_global_load_async_to_lds_b32)
  #define HAVE_ASYNC_LDS 1
// Probe-confirmed: param 0 is non-const AS(1) int*, param 1 AS(3) int*
typedef __attribute__((address_space(1))) int g1_int;
typedef __attribute__((address_space(3))) int l3_int;
#endif

__device__ __forceinline__ void wait_asynccnt0() {
#if defined(HAVE_ASYNC_LDS)
  #if __has_builtin(__builtin_amdgcn_s_wait_asynccnt)
  __builtin_amdgcn_s_wait_asynccnt(0);
  #else
  asm volatile("s_wait_asynccnt 0x0" ::: "memory");
  #endif
#endif
}

// ---------------- zero workspace ----------------
__global__ void nn_zero_kernel(float* __restrict__ p, int n) {
  int i = blockIdx.x * blockDim.x + threadIdx.x;
  if (i < n) p[i] = 0.0f;
}

// ---------------- edge scatter: acc[i][f] += feat[j][f]; optional deg[i]++ ----
// One wave handles 16 edges: lanes 0-15 stage idx_i, lanes 16-31 stage idx_j
// into LDS via ASYNC loads (ASYNCcnt path), then loop edges with a coalesced
// 128B gather of the source row and 32 contiguous f32 atomics
// (global_atomic_add_f32, no-return -> STOREcnt) resolving in L2
// (all accumulators are L2-resident: ~13 MB total vs 192 MB L2).
template <bool ADD_DEG>
__global__ void nn_scatter_kernel(const float* __restrict__ feat,
                                  const int* __restrict__ idx_i,
                                  const int* __restrict__ idx_j,
                                  float* __restrict__ acc,
                                  float* __restrict__ deg,
                                  int nE) {
  constexpr int EPW = 16;               // edges per wave
  __shared__ int sIdx[8 * 2 * EPW];     // 8 waves per 256-thread block

  const int lane = threadIdx.x & 31;
  const int wib  = threadIdx.x >> 5;
  const long wave = (long)blockIdx.x * (blockDim.x >> 5) + wib;
  const long e0 = wave * EPW;
  if (e0 >= nE) return;
  const int nEdges = (nE - e0 < (long)EPW) ? (int)(nE - e0) : EPW;

  int* li = &sIdx[wib * 2 * EPW];
  int* lj = li + EPW;

  const int t = lane & 15;
  const bool isI = lane < 16;
  if (t < nEdges) {
#if defined(HAVE_ASYNC_LDS)
    const int* g = isI ? (idx_i + e0 + t) : (idx_j + e0 + t);
    int* l = isI ? (li + t) : (lj + t);
    __builtin_amdgcn_global_load_async_to_lds_b32((g1_int*)g, (l3_int*)l, 0, 0);
#else
    if (isI) li[t] = idx_i[e0 + t];
    else     lj[t] = idx_j[e0 + t];
#endif
  }
#if defined(HAVE_ASYNC_LDS)
  wait_asynccnt0();
#endif
  // li/lj written by lanes of this same wave; DS ops are in-order per wave,
  // and s_wait_asynccnt 0 covers all of this wave's async LDS writes.

  for (int k = 0; k < nEdges; ++k) {
    const int i = li[k];
    const int j = lj[k];
    if (k + 1 < nEdges) {
      // software pipeline: prefetch next source row while this atomic drains
      __builtin_prefetch(&feat[(size_t)lj[k + 1] * DF + lane], 0, 0);
    }
    const float v = feat[(size_t)j * DF + lane];
    unsafeAtomicAdd(&acc[(size_t)i * DF + lane], v);
    if (ADD_DEG && lane == 0) unsafeAtomicAdd(deg + i, 1.0f);
  }
}

// ---------------- mean + squared-deviation kernel ----------------
// Per wave: a 16-node x 32-feature tile. mean = diag(1/deg) * SumTile is a
// matmul -> chain of 4x v_wmma_f32_16x16x4_f32 per 16x16 tile.
// A (16x4, documented layout): lane L holds row m = L&15, regs = K {0,1} for
// lanes 0-15 and K {2,3} for lanes 16-31. A_b[m,k] = (4b+k==m) ? 1/deg[m] : 0.
// B (4x16, layout by analogy to C/D): reg r: lanes 0-15 -> K=r, N=lane;
// lanes 16-31 -> K=r+2, N=lane-16. B_b[k,n] = Sum[node0+4b+k][f0+n].
// D (16x16, documented): reg d: lanes 0-15 -> M=d, N=lane; 16-31 -> M=d+8.
// Epilogue: rx = (x - mean)^2 written in place over the sum buffer (each tile
// is owned by exactly one wave; reads precede writes within the wave). The
// full-tile case (always, for N % 16 == 0) takes a uniform branch with
// unguarded coalesced stores — no per-element EXEC toggling.
__global__ void nn_meansq_kernel(const float* __restrict__ x,
                                 const float* __restrict__ deg,
                                 const float* __restrict__ sum,
                                 float* __restrict__ rx,
                                 int nN) {
  const int lane = threadIdx.x & 31;
  const long wave = ((long)blockIdx.x * blockDim.x + threadIdx.x) >> 5;
  const int node0 = (int)(wave * 16);
  if (node0 >= nN) return;
  const int l15 = lane & 15;
  const int half = lane >> 4;
  const bool full = (node0 + 16 <= nN);   // wave-uniform

  const int nodeL = min(node0 + l15, nN - 1);
  const float rinv = 1.0f / fmaxf(deg[nodeL], 1.0f);

#if defined(HAVE_WMMA_F32)
  const int K0 = 2 * half;
  for (int f0 = 0; f0 < DF; f0 += 16) {
    v8f accv = {};
#pragma unroll
    for (int b = 0; b < 4; ++b) {
      v2f a, bb;
      a.x = (4 * b + K0 + 0 == l15) ? rinv : 0.0f;
      a.y = (4 * b + K0 + 1 == l15) ? rinv : 0.0f;
      const int r0 = min(node0 + 4 * b + K0 + 0, nN - 1);
      const int r1 = min(node0 + 4 * b + K0 + 1, nN - 1);
      bb.x = sum[(size_t)r0 * DF + f0 + l15];
      bb.y = sum[(size_t)r1 * DF + f0 + l15];
      accv = __builtin_amdgcn_wmma_f32_16x16x4_f32(false, a, false, bb,
                                                   (short)0, accv, false, false);
    }
    if (full) {
#pragma unroll
      for (int d = 0; d < 8; ++d) {
        const size_t addr = (size_t)(node0 + d + 8 * half) * DF + f0 + l15;
        const float dv = x[addr] - accv[d];
        rx[addr] = dv * dv;
      }
    } else {
#pragma unroll
      for (int d = 0; d < 8; ++d) {
        const int node = node0 + d + 8 * half;
        if (node < nN) {
          const size_t addr = (size_t)node * DF + f0 + l15;
          const float dv = x[addr] - accv[d];
          rx[addr] = dv * dv;
        }
      }
    }
  }
#else
  for (int f0 = 0; f0 < DF; f0 += 16) {
    for (int d = 0; d < 8; ++d) {
      const int node = node0 + d + 8 * half;
      if (node < nN) {
        const size_t addr = (size_t)node * DF + f0 + l15;
        const float ri = 1.0f / fmaxf(deg[node], 1.0f);
        const float dv = x[addr] - sum[addr] * ri;
        rx[addr] = dv * dv;
      }
    }
  }
#endif
}

// ---------------- finalize: var = diag(1/deg)*VarSum; out = x/sqrt(var+eps) --
__global__ void nn_finalize_kernel(const float* __restrict__ x,
                                   const float* __restrict__ deg,
                                   const float* __restrict__ varsum,
                                   float* __restrict__ out,
                                   int nN) {
  const int lane = threadIdx.x & 31;
  const long wave = ((long)blockIdx.x * blockDim.x + threadIdx.x) >> 5;
  const int node0 = (int)(wave * 16);
  if (node0 >= nN) return;
  const int l15 = lane & 15;
  const int half = lane >> 4;
  const bool full = (node0 + 16 <= nN);   // wave-uniform

  const int nodeL = min(node0 + l15, nN - 1);
  const float rinv = 1.0f / fmaxf(deg[nodeL], 1.0f);

#if defined(HAVE_WMMA_F32)
  const int K0 = 2 * half;
  for (int f0 = 0; f0 < DF; f0 += 16) {
    v8f accv = {};
#pragma unroll
    for (int b = 0; b < 4; ++b) {
      v2f a, bb;
      a.x = (4 * b + K0 + 0 == l15) ? rinv : 0.0f;
      a.y = (4 * b + K0 + 1 == l15) ? rinv : 0.0f;
      const int r0 = min(node0 + 4 * b + K0 + 0, nN - 1);
      const int r1 = min(node0 + 4 * b + K0 + 1, nN - 1);
      bb.x = varsum[(size_t)r0 * DF + f0 + l15];
      bb.y = varsum[(size_t)r1 * DF + f0 + l15];
      accv = __builtin_amdgcn_wmma_f32_16x16x4_f32(false, a, false, bb,
                                                   (short)0, accv, false, false);
    }
    if (full) {
#pragma unroll
      for (int d = 0; d < 8; ++d) {
        const size_t addr = (size_t)(node0 + d + 8 * half) * DF + f0 + l15;
        const float xv = x[addr];
        const float sd = sqrtf(accv[d] + EPSF);
        float o = xv / sd;
        if (__builtin_isinf(o)) o = xv;
        out[addr] = o;
      }
    } else {
#pragma unroll
      for (int d = 0; d < 8; ++d) {
        const int node = node0 + d + 8 * half;
        if (node < nN) {
          const size_t addr = (size_t)node * DF + f0 + l15;
          const float xv = x[addr];
          const float sd = sqrtf(accv[d] + EPSF);
          float o = xv / sd;
          if (__builtin_isinf(o)) o = xv;
          out[addr] = o;
        }
      }
    }
  }
#else
  for (int f0 = 0; f0 < DF; f0 += 16) {
    for (int d = 0; d < 8; ++d) {
      const int node = node0 + d + 8 * half;
      if (node < nN) {
        const size_t addr = (size_t)node * DF + f0 + l15;
        const float ri = 1.0f / fmaxf(deg[node], 1.0f);
        const float xv = x[addr];
        const float sd = sqrtf(varsum[addr] * ri + EPSF);
        float o = xv / sd;
        if (__builtin_isinf(o)) o = xv;
        out[addr] = o;
      }
    }
  }
#endif
}

// ---------------- launcher ----------------
extern "C" void kernel_launch(void* const* d_in, const int* in_sizes, int n_in,
                              void* d_out, int out_size, void* d_ws, size_t ws_size,
                              hipStream_t stream) {
  const float* x = (const float*)d_in[0];
  const int* ei  = (const int*)d_in[1];
  const int nN = in_sizes[0] / DF;
  const int nE = in_sizes[1] / 2;
  const int* idx_i = ei;        // targets (segments)
  const int* idx_j = ei + nE;   // sources (gathered)
  float* out = (float*)d_out;

  // workspace layout: deg[nN] | sum/rx[nN*DF] | varsum[nN*DF]  (~13 MB)
  float* deg    = (float*)d_ws;
  float* sum    = deg + nN;                 // reused in-place as rx
  float* varsum = sum + (size_t)nN * DF;

  const int zn = nN * (1 + 2 * DF);
  nn_zero_kernel<<<(zn + 255) / 256, 256, 0, stream>>>(deg, zn);

  const long waves_e  = ((long)nE + 15) / 16;
  const int  blocks_e = (int)((waves_e + 7) / 8);          // 8 waves / block
  nn_scatter_kernel<true><<<blocks_e, 256, 0, stream>>>(x, idx_i, idx_j,
                                                        sum, deg, nE);

  const long waves_n  = ((long)nN + 15) / 16;
  const int  blocks_n = (int)((waves_n * 32 + 255) / 256);
  nn_meansq_kernel<<<blocks_n, 256, 0, stream>>>(x, deg, sum, sum, nN);

  nn_scatter_kernel<false><<<blocks_e, 256, 0, stream>>>(sum /*rx*/, idx_i, idx_j,
                                                         varsum, deg, nE);

  nn_finalize_kernel<<<blocks_n, 256, 0, stream>>>(x, deg, varsum, out, nN);
}